// WalletGNN_10161892622477
// MI455X (gfx1250) — compile-verified
//
#include <hip/hip_runtime.h>
#include <hip/hip_bf16.h>

typedef __attribute__((ext_vector_type(2))) float v2f;
typedef __attribute__((ext_vector_type(8))) float v8f;

#define DIN 128   // input dim of both layers' GEMMs (din = dh = 128)

// ---------------------------------------------------------------------------
// degree / normalization kernels
// ---------------------------------------------------------------------------
__global__ __launch_bounds__(256) void deg_init_kernel(float* deg, int n) {
    int i = blockIdx.x * blockDim.x + threadIdx.x;
    if (i < n) deg[i] = 1.0f;  // self-loop contributes 1 to every dst degree
}

__global__ __launch_bounds__(256) void deg_accum_kernel(const int* __restrict__ ei, int e,
                                                        float* __restrict__ deg) {
    int idx = blockIdx.x * blockDim.x + threadIdx.x;
    if (idx < e) atomicAdd(&deg[ei[e + idx]], 1.0f);  // dst row of edge_index
}

__global__ __launch_bounds__(256) void deg_to_dinv_kernel(float* deg, int n) {
    int i = blockIdx.x * blockDim.x + threadIdx.x;
    if (i < n) deg[i] = rsqrtf(deg[i]);  // deg >= 1 always (self loops)
}

// ---------------------------------------------------------------------------
// WMMA fp32 GEMM:  H[N x DOUT] = X[N x 128] * W[128 x DOUT]
// 256 threads = 8 wave32 per block. W is staged into LDS PRE-SWIZZLED into
// B-fragment order so each lane's (b.x,b.y) pair is one 8-byte ds_load_b64
// straight into an even-aligned VGPR pair (no shuffle movs before WMMA).
// Each wave owns a 16-row strip and accumulates DOUT/16 16x16 f32 tiles with
// v_wmma_f32_16x16x4_f32 over 32 k-steps.
// ---------------------------------------------------------------------------
template <int DOUT>
__global__ __launch_bounds__(256) void gemm_wmma_kernel(const float* __restrict__ X,
                                                        const float* __restrict__ W,
                                                        float* __restrict__ H, int n) {
    constexpr int NT = DOUT / 16;          // 16x16 col tiles per wave
    constexpr int NPAIR = 32 * NT * 32;    // 32 k-steps * NT tiles * 32 lanes
    __shared__ float sW[NPAIR * 2];        // 64 KB (DOUT=128) / 32 KB (DOUT=64)

    // Stage + swizzle W into per-fragment layout:
    //   sW[((ks*NT + t)*32 + l)*2 + j] = W[(4*ks + (l>>4)*2 + j)*DOUT + t*16 + (l&15)]
    for (int pair = threadIdx.x; pair < NPAIR; pair += 256) {
        int l  = pair & 31;
        int tt = (pair >> 5) % NT;
        int ks = (pair >> 5) / NT;
        int col = tt * 16 + (l & 15);
        int k0  = ks * 4 + ((l >> 4) * 2);
        sW[pair * 2 + 0] = W[k0 * DOUT + col];
        sW[pair * 2 + 1] = W[(k0 + 1) * DOUT + col];
    }
    __syncthreads();

    const int wave = threadIdx.x >> 5;
    const int lane = threadIdx.x & 31;
    const int m    = lane & 15;        // row-in-tile (A) / col-in-tile (B,C)
    const int kk   = (lane >> 4) * 2;  // K sub-offset: lanes 0-15 -> {0,1}, 16-31 -> {2,3}

    const long long row0 = (long long)blockIdx.x * 128 + wave * 16;
    long long rowm = row0 + m;
    if (rowm >= n) rowm = n - 1;       // clamp loads; stores are guarded below
    const float* __restrict__ xrow = X + rowm * DIN + kk;
    const float* __restrict__ bl   = sW + lane * 2;  // per-lane fragment base

    v8f acc[NT] = {};

#pragma unroll 4
    for (int ks = 0; ks < 32; ks++) {
        v2f a = *(const v2f*)(xrow + ks * 4);   // 8B-aligned global_load_b64
#pragma unroll
        for (int t = 0; t < NT; t++) {
            v2f b = *(const v2f*)(bl + (ks * NT + t) * 64);  // ds_load_b64, imm offset
            acc[t] = __builtin_amdgcn_wmma_f32_16x16x4_f32(
                false, a, false, b, (short)0, acc[t], false, false);
        }
    }

    // C/D layout: VGPR v, lanes 0-15 -> row M=v; lanes 16-31 -> row M=v+8
    const long long rbase = row0 + (lane >> 4) * 8;
#pragma unroll
    for (int v = 0; v < 8; v++) {
        long long r = rbase + v;
        if (r < n) {
            float* __restrict__ orow = H + r * DOUT;
#pragma unroll
            for (int t = 0; t < NT; t++) orow[t * 16 + m] = acc[t][v];
        }
    }
}

// ---------------------------------------------------------------------------
// Agg init: agg[i,:] = bias + h[i,:] * dinv[i]^2   (self-loop term folded in)
// ---------------------------------------------------------------------------
template <int D>
__global__ __launch_bounds__(256) void init_agg_kernel(const float* __restrict__ H,
                                                       const float* __restrict__ bias,
                                                       const float* __restrict__ dinv,
                                                       float* __restrict__ Agg, int n) {
    long long idx = (long long)blockIdx.x * blockDim.x + threadIdx.x;
    long long total = (long long)n * D;
    if (idx < total) {
        int r = (int)(idx / D);
        int c = (int)(idx - (long long)r * D);
        float w = dinv[r];
        Agg[idx] = bias[c] + H[idx] * (w * w);
    }
}

// ---------------------------------------------------------------------------
// Edge scatter: one wave32 per edge; lane handles D/32 consecutive floats.
// agg[dst,:] += h[src,:] * (dinv[src]*dinv[dst])   via f32 global atomics.
// ---------------------------------------------------------------------------
template <int D>
__global__ __launch_bounds__(256) void scatter_kernel(const int* __restrict__ ei, int e,
                                                      const float* __restrict__ dinv,
                                                      const float* __restrict__ H,
                                                      float* __restrict__ Agg) {
    constexpr int VEC = D / 32;  // 4 (D=128) or 2 (D=64)
    const int lane = threadIdx.x & 31;
    const int wid  = (int)((blockIdx.x * (long long)blockDim.x + threadIdx.x) >> 5);
    const int nw   = (int)(((long long)gridDim.x * blockDim.x) >> 5);

    for (int ed = wid; ed < e; ed += nw) {
        int s = ei[ed];
        int d = ei[e + ed];
        float w = dinv[s] * dinv[d];

        // prefetch next edge's source row (L2/HBM latency hiding)
        int e2 = ed + nw;
        if (e2 < e) {
            int s2 = ei[e2];
            __builtin_prefetch(H + (long long)s2 * D + lane * VEC, 0, 1);
        }

        const float* __restrict__ hs = H + (long long)s * D + lane * VEC;
        float* __restrict__ ad = Agg + (long long)d * D + lane * VEC;
        if (VEC == 4) {
            float4 v = *(const float4*)hs;
            atomicAdd(ad + 0, v.x * w);
            atomicAdd(ad + 1, v.y * w);
            atomicAdd(ad + 2, v.z * w);
            atomicAdd(ad + 3, v.w * w);
        } else {
            float2 v = *(const float2*)hs;
            atomicAdd(ad + 0, v.x * w);
            atomicAdd(ad + 1, v.y * w);
        }
    }
}

// ---------------------------------------------------------------------------
// ELU in place (alpha = 1)
// ---------------------------------------------------------------------------
__global__ __launch_bounds__(256) void elu_kernel(float* __restrict__ A, long long total) {
    long long idx = (long long)blockIdx.x * blockDim.x + threadIdx.x;
    if (idx < total) {
        float v = A[idx];
        A[idx] = v > 0.0f ? v : (__expf(v) - 1.0f);
    }
}

// ---------------------------------------------------------------------------
// launch
// ---------------------------------------------------------------------------
extern "C" void kernel_launch(void* const* d_in, const int* in_sizes, int n_in,
                              void* d_out, int out_size, void* d_ws, size_t ws_size,
                              hipStream_t stream) {
    const float* x  = (const float*)d_in[0];
    const int*   ei = (const int*)d_in[1];     // [2, E] row-major: src row then dst row
    const float* W1 = (const float*)d_in[2];   // [128,128]
    const float* b1 = (const float*)d_in[3];   // [128]
    const float* W2 = (const float*)d_in[4];   // [128,64]
    const float* b2 = (const float*)d_in[5];   // [64]
    float* out = (float*)d_out;                // [N,64]

    const int N = in_sizes[0] / DIN;
    const int E = in_sizes[1] / 2;
    const int DH = 128, DOUT = 64;

    // scratch layout: dinv | hbuf (N*128) | aggbuf (N*128)
    char* ws = (char*)d_ws;
    float* dinv = (float*)ws;
    size_t off = (((size_t)N * sizeof(float)) + 255) & ~(size_t)255;
    float* hbuf = (float*)(ws + off);
    float* aggbuf = (float*)(ws + off + (size_t)N * DH * sizeof(float));

    const int B = 256;
    const int gN    = (N + B - 1) / B;
    const int gE    = (E + B - 1) / B;
    const int gRows = (N + 127) / 128;                       // 128 rows per block (8 waves x 16)
    const int gNDH  = (int)(((long long)N * DH + B - 1) / B);
    const int gNDO  = (int)(((long long)N * DOUT + B - 1) / B);
    const int gEdgeW = (E + 7) / 8;                          // 8 waves per block, 1 edge per wave

    // --- normalization ---
    deg_init_kernel<<<gN, B, 0, stream>>>(dinv, N);
    deg_accum_kernel<<<gE, B, 0, stream>>>(ei, E, dinv);
    deg_to_dinv_kernel<<<gN, B, 0, stream>>>(dinv, N);

    // --- layer 1: h = x @ W1 ; agg = b1 + self + scatter ; elu ---
    gemm_wmma_kernel<128><<<gRows, B, 0, stream>>>(x, W1, hbuf, N);
    init_agg_kernel<128><<<gNDH, B, 0, stream>>>(hbuf, b1, dinv, aggbuf, N);
    scatter_kernel<128><<<gEdgeW, B, 0, stream>>>(ei, E, dinv, hbuf, aggbuf);
    elu_kernel<<<gNDH, B, 0, stream>>>(aggbuf, (long long)N * DH);

    // --- layer 2: h2 = g @ W2 (reuse hbuf) ; out = b2 + self + scatter ---
    gemm_wmma_kernel<64><<<gRows, B, 0, stream>>>(aggbuf, W2, hbuf, N);
    init_agg_kernel<64><<<gNDO, B, 0, stream>>>(hbuf, b2, dinv, out, N);
    scatter_kernel<64><<<gEdgeW, B, 0, stream>>>(ei, E, dinv, hbuf, out);
}